// Smk3dLoss_85658827751823
// MI455X (gfx1250) — compile-verified
//
#include <hip/hip_runtime.h>
#include <hip/hip_bf16.h>
#include <math.h>

#define PI_F 3.14159265358979323846f

typedef float v2f __attribute__((ext_vector_type(2)));
typedef float v8f __attribute__((ext_vector_type(8)));

// Full-wave (32-lane) sum using the CDNA5 matrix pipe.
// A (16x4 f32): lanes 0-15 hold rows 0-15; VGPR0 = K0 (lanes 0-15) / K2 (lanes 16-31),
// VGPR1 = K1/K3.  We place the per-lane partial in VGPR0, zero in VGPR1.
// B = all-ones (layout independent), so D[m][n] = v[m] + v[m+16] for every n.
// D layout: lane L<16 holds D[0..7][L], lane L>=16 holds D[8..15][L-16]; summing the
// 8 accumulator regs per lane gives half-sums, one shfl_xor(16) completes the total
// (broadcast to all lanes).  Requires EXEC == all ones: only call with a full wave.
__device__ __forceinline__ float wmma_wave_sum(float v) {
    v2f a; a[0] = v;    a[1] = 0.0f;
    v2f b; b[0] = 1.0f; b[1] = 1.0f;
    v8f c = {};
    c = __builtin_amdgcn_wmma_f32_16x16x4_f32(false, a, false, b,
                                              (short)0, c, false, false);
    float s = c[0] + c[1] + c[2] + c[3] + c[4] + c[5] + c[6] + c[7];
    s += __shfl_xor(s, 16, 32);
    return s;
}

// ---------------------------------------------------------------------------
// Kernel 1: focal loss partial reduction over hm/hm_gt (the bandwidth-bound part)
// Each block consumes 2048 floats (512 float4) from each array -> 2880 blocks.
// ---------------------------------------------------------------------------
__global__ __launch_bounds__(256) void focal_partial(
    const float* __restrict__ hm, const float* __restrict__ gt,
    float* __restrict__ outP, int n4) {
    const int tid = threadIdx.x;
    const long base = (long)blockIdx.x * 512;
    const float4* __restrict__ hm4 = (const float4*)hm;
    const float4* __restrict__ gt4 = (const float4*)gt;

    float pl = 0.f, nl = 0.f, np = 0.f;
#pragma unroll
    for (int it = 0; it < 2; ++it) {
        long i = base + it * 256 + tid;
        if (i < n4) {
            float4 x = hm4[i];
            float4 g = gt4[i];
            float xs[4] = {x.x, x.y, x.z, x.w};
            float gs[4] = {g.x, g.y, g.z, g.w};
#pragma unroll
            for (int j = 0; j < 4; ++j) {
                float p = __fdividef(1.0f, 1.0f + __expf(-xs[j]));
                p = fminf(fmaxf(p, 1e-4f), 1.0f - 1e-4f);
                float gv = gs[j];
                float omp = 1.0f - p;
                if (gv == 1.0f) {
                    pl += __logf(p) * omp * omp;
                    np += 1.0f;
                } else {
                    float ng  = 1.0f - gv;
                    float ng2 = ng * ng;
                    nl += __logf(omp) * p * p * (ng2 * ng2);
                }
            }
        }
    }

    __shared__ float lds[8 * 3];
    const int lane = tid & 31, wave = tid >> 5;
    pl = wmma_wave_sum(pl);
    nl = wmma_wave_sum(nl);
    np = wmma_wave_sum(np);
    if (lane == 0) {
        lds[wave * 3 + 0] = pl;
        lds[wave * 3 + 1] = nl;
        lds[wave * 3 + 2] = np;
    }
    __syncthreads();
    if (wave == 0) {
        float a0 = (lane < 8) ? lds[lane * 3 + 0] : 0.f;
        float a1 = (lane < 8) ? lds[lane * 3 + 1] : 0.f;
        float a2 = (lane < 8) ? lds[lane * 3 + 2] : 0.f;
        a0 = wmma_wave_sum(a0);
        a1 = wmma_wave_sum(a1);
        a2 = wmma_wave_sum(a2);
        if (lane == 0) {
            outP[blockIdx.x * 3 + 0] = a0;
            outP[blockIdx.x * 3 + 1] = a1;
            outP[blockIdx.x * 3 + 2] = a2;
        }
    }
}

// ---------------------------------------------------------------------------
// Kernel 2: per-object SMOKE decode + disentangled L1 partials (4096 objects)
// ---------------------------------------------------------------------------
__device__ __forceinline__ void inv3(const float* __restrict__ M, float* __restrict__ o) {
    float a = M[0], b = M[1], c = M[2];
    float d = M[3], e = M[4], f = M[5];
    float g = M[6], h = M[7], i = M[8];
    float A  = e * i - f * h;
    float Bv = f * g - d * i;
    float Cv = d * h - e * g;
    float det = a * A + b * Bv + c * Cv;
    float inv = 1.0f / det;
    o[0] = A * inv;             o[1] = (c * h - b * i) * inv;  o[2] = (b * f - c * e) * inv;
    o[3] = Bv * inv;            o[4] = (a * i - c * g) * inv;  o[5] = (c * d - a * f) * inv;
    o[6] = Cv * inv;            o[7] = (b * g - a * h) * inv;  o[8] = (a * e - b * d) * inv;
}

// Sum_j | corner_j(roty, dims, locs) - reg_j |, corners flattened [3][8]
__device__ __forceinline__ float box_l1(float roty, const float* D, const float* L,
                                        const float* __restrict__ rg) {
    float c, s;
    __sincosf(roty, &s, &c);
    const float sx[8] = {-.5f, .5f, .5f, .5f, .5f, -.5f, -.5f, -.5f};
    const float sy[8] = {-1.f, -1.f, 0.f, 0.f, -1.f, -1.f, 0.f, 0.f};
    const float sz[8] = {-.5f, -.5f, -.5f, .5f, .5f, .5f, .5f, -.5f};
    float acc = 0.f;
#pragma unroll
    for (int i = 0; i < 8; ++i) {
        float x = D[0] * sx[i];
        float y = D[1] * sy[i];
        float z = D[2] * sz[i];
        float X =  c * x + s * z + L[0];
        float Y =  y + L[1];
        float Z = -s * x + c * z + L[2];
        acc += fabsf(X - rg[i]) + fabsf(Y - rg[8 + i]) + fabsf(Z - rg[16 + i]);
    }
    return acc;
}

__global__ __launch_bounds__(256) void box_partial(
    const float* __restrict__ dep, const float* __restrict__ offset,
    const float* __restrict__ dimf, const float* __restrict__ rot,
    const float* __restrict__ reg, const float* __restrict__ proj_cts,
    const float* __restrict__ K_mat, const float* __restrict__ trans_mat,
    const float* __restrict__ locs, const float* __restrict__ dims_gt,
    const float* __restrict__ rotys_gt, const int* __restrict__ ind,
    const int* __restrict__ reg_mask, const int* __restrict__ cls_ids,
    float* __restrict__ outP, int nobj) {
    const int tid = threadIdx.x;
    const int obj = blockIdx.x * 256 + tid;
    float s_ori = 0.f, s_dim = 0.f, s_loc = 0.f, cnt = 0.f;

    if (obj < nobj) {
        const int HW = 96 * 320;
        const int b  = obj >> 6;
        const int id = ind[obj];
        const int m  = reg_mask[obj];

        // gather predictions at gt center index
        float dep_o = dep[b * HW + id];
        float off0  = offset[b * 2 * HW + id];
        float off1  = offset[b * 2 * HW + HW + id];
        float d0 = dimf[b * 3 * HW + id];
        float d1 = dimf[b * 3 * HW + HW + id];
        float d2 = dimf[b * 3 * HW + 2 * HW + id];
        float r0 = rot[b * 2 * HW + id];
        float r1 = rot[b * 2 * HW + HW + id];

        float depth = dep_o * 16.32f + 28.01f;
        const int cid = cls_ids[obj];
        const float DR[3][3] = {{3.88f, 1.63f, 1.53f},
                                {1.78f, 1.70f, 0.58f},
                                {0.88f, 1.73f, 0.67f}};
        float pd[3] = {__expf(d0) * DR[cid][0],
                       __expf(d1) * DR[cid][1],
                       __expf(d2) * DR[cid][2]};

        float ptsx = proj_cts[obj * 2 + 0] + off0;
        float ptsy = proj_cts[obj * 2 + 1] + off1;

        float Ti[9], Ki[9];
        inv3(trans_mat + b * 9, Ti);
        inv3(K_mat + b * 9, Ki);

        float img0 = (Ti[0] * ptsx + Ti[1] * ptsy + Ti[2]) * depth;
        float img1 = (Ti[3] * ptsx + Ti[4] * ptsy + Ti[5]) * depth;
        float img2 = (Ti[6] * ptsx + Ti[7] * ptsy + Ti[8]) * depth;
        float pl3[3];
        pl3[0] = Ki[0] * img0 + Ki[1] * img1 + Ki[2] * img2;
        pl3[1] = Ki[3] * img0 + Ki[4] * img1 + Ki[5] * img2 + 0.5f * pd[1];
        pl3[2] = Ki[6] * img0 + Ki[7] * img1 + Ki[8] * img2;

        const float* L = locs + obj * 3;
        float rays  = atanf(L[0] / (L[2] + 1e-7f));
        float alpha = atanf(r0 / (r1 + 1e-7f)) +
                      ((r1 >= 0.f) ? -PI_F * 0.5f : PI_F * 0.5f);
        float roty = alpha + rays;
        if (roty >  PI_F) roty -= 2.f * PI_F;
        if (roty < -PI_F) roty += 2.f * PI_F;

        const float* Dg = dims_gt + obj * 3;
        const float* rg = reg + obj * 24;
        float rgy = rotys_gt[obj];
        if (m) {
            s_ori = box_l1(roty, Dg, L, rg);
            s_dim = box_l1(rgy, pd, L, rg);
            s_loc = box_l1(rgy, Dg, pl3, rg);
            cnt = 1.0f;
        }
    }

    __shared__ float lds[8 * 4];
    const int lane = tid & 31, wave = tid >> 5;
    s_ori = wmma_wave_sum(s_ori);
    s_dim = wmma_wave_sum(s_dim);
    s_loc = wmma_wave_sum(s_loc);
    cnt   = wmma_wave_sum(cnt);
    if (lane == 0) {
        lds[wave * 4 + 0] = s_ori;
        lds[wave * 4 + 1] = s_dim;
        lds[wave * 4 + 2] = s_loc;
        lds[wave * 4 + 3] = cnt;
    }
    __syncthreads();
    if (wave == 0) {
        float a0 = (lane < 8) ? lds[lane * 4 + 0] : 0.f;
        float a1 = (lane < 8) ? lds[lane * 4 + 1] : 0.f;
        float a2 = (lane < 8) ? lds[lane * 4 + 2] : 0.f;
        float a3 = (lane < 8) ? lds[lane * 4 + 3] : 0.f;
        a0 = wmma_wave_sum(a0);
        a1 = wmma_wave_sum(a1);
        a2 = wmma_wave_sum(a2);
        a3 = wmma_wave_sum(a3);
        if (lane == 0) {
            outP[blockIdx.x * 4 + 0] = a0;
            outP[blockIdx.x * 4 + 1] = a1;
            outP[blockIdx.x * 4 + 2] = a2;
            outP[blockIdx.x * 4 + 3] = a3;
        }
    }
}

// ---------------------------------------------------------------------------
// Kernel 3: fold partials, emit the 6 scalar losses
// ---------------------------------------------------------------------------
__global__ __launch_bounds__(256) void combine_losses(
    const float* __restrict__ focalP, int nfb,
    const float* __restrict__ boxP, int nbb,
    float* __restrict__ out) {
    const int tid = threadIdx.x;
    float acc[7] = {0.f, 0.f, 0.f, 0.f, 0.f, 0.f, 0.f};
    for (int i = tid; i < nfb; i += 256) {
        acc[0] += focalP[i * 3 + 0];
        acc[1] += focalP[i * 3 + 1];
        acc[2] += focalP[i * 3 + 2];
    }
    for (int i = tid; i < nbb; i += 256) {
        acc[3] += boxP[i * 4 + 0];
        acc[4] += boxP[i * 4 + 1];
        acc[5] += boxP[i * 4 + 2];
        acc[6] += boxP[i * 4 + 3];
    }

    __shared__ float lds[8 * 7];
    const int lane = tid & 31, wave = tid >> 5;
#pragma unroll
    for (int q = 0; q < 7; ++q) acc[q] = wmma_wave_sum(acc[q]);
    if (lane == 0) {
#pragma unroll
        for (int q = 0; q < 7; ++q) lds[wave * 7 + q] = acc[q];
    }
    __syncthreads();
    if (wave == 0) {
#pragma unroll
        for (int q = 0; q < 7; ++q) {
            float x = (lane < 8) ? lds[lane * 7 + q] : 0.f;
            acc[q] = wmma_wave_sum(x);
        }
        if (lane == 0) {
            float pl = acc[0], nl = acc[1], np = acc[2];
            float hm_loss = (np == 0.f) ? -nl : -(pl + nl) / fmaxf(np, 1.f);
            float denom = 24.f * acc[6] + 1e-4f;   // mask expanded over 24 comps
            float ori = acc[3] / denom;
            float dml = acc[4] / denom;
            float loc = acc[5] / denom;
            float reg_loss = ori + 2.f * dml + loc;  // ORI_W=1, DIM_W=2, LOC_W=1
            out[0] = hm_loss + reg_loss;
            out[1] = hm_loss;
            out[2] = reg_loss;
            out[3] = ori;
            out[4] = dml;
            out[5] = loc;
        }
    }
}

extern "C" void kernel_launch(void* const* d_in, const int* in_sizes, int n_in,
                              void* d_out, int out_size, void* d_ws, size_t ws_size,
                              hipStream_t stream) {
    const float* hm        = (const float*)d_in[0];
    const float* dep       = (const float*)d_in[1];
    const float* offset    = (const float*)d_in[2];
    const float* dimf      = (const float*)d_in[3];
    const float* rot       = (const float*)d_in[4];
    const float* hm_gt     = (const float*)d_in[5];
    const float* reg       = (const float*)d_in[6];
    const float* proj_cts  = (const float*)d_in[7];
    const float* K_mat     = (const float*)d_in[8];
    const float* trans_mat = (const float*)d_in[9];
    const float* locs      = (const float*)d_in[10];
    const float* dims_gt   = (const float*)d_in[11];
    const float* rotys_gt  = (const float*)d_in[12];
    const int*   ind       = (const int*)d_in[13];
    const int*   reg_mask  = (const int*)d_in[14];
    const int*   cls_ids   = (const int*)d_in[15];

    float* out = (float*)d_out;
    float* ws  = (float*)d_ws;

    const int NHM = in_sizes[0];             // B*C*H*W = 5,898,240
    const int n4  = NHM / 4;                 // float4 count
    const int nfb = (NHM + 2047) / 2048;     // 2880 focal blocks
    const int NOBJ = in_sizes[13];           // B*K = 4096
    const int nbb  = (NOBJ + 255) / 256;     // 16 box blocks

    float* focalP = ws;                      // nfb*3 floats
    float* boxP   = ws + (size_t)nfb * 3;    // nbb*4 floats

    focal_partial<<<nfb, 256, 0, stream>>>(hm, hm_gt, focalP, n4);
    box_partial<<<nbb, 256, 0, stream>>>(dep, offset, dimf, rot, reg, proj_cts,
                                         K_mat, trans_mat, locs, dims_gt,
                                         rotys_gt, ind, reg_mask, cls_ids,
                                         boxP, NOBJ);
    combine_losses<<<1, 256, 0, stream>>>(focalP, nfb, boxP, nbb, out);
}